// HybridNetBackbone_74852690034955
// MI455X (gfx1250) — compile-verified
//
#include <hip/hip_runtime.h>
#include <math.h>

#define BB 2
#define NV 6
#define NJ 23
#define HM 128
#define W_PAD 640
#define H_PAD 512
#define G 52
#define GP 54      // padded x,y dims of Vol
#define GZ 72      // padded z dim of Vol (room for z0+15+kw)
#define KTOT 621   // 23*27
#define KPAD 624   // padded to multiple of 4
#define CO_PAD 32  // 23 out channels padded to 2x16

typedef float v8f __attribute__((ext_vector_type(8)));
typedef float v2f __attribute__((ext_vector_type(2)));

static const long long VOL_CI = (long long)GP * GP * GZ;          // 209952
static const long long VOL_B  = (long long)NJ * GP * GP * GZ;     // 4828896
static const long long VOL_N  = (long long)BB * NJ * GP * GP * GZ;
static const long long HF_N   = (long long)BB * NJ * G * G * G;   // 6467968
static const long long PAD_N  = (long long)BB * NV * NJ * H_PAD * W_PAD; // 90439680

// ---- 1) padded heatmap scatter: write the `padded` output directly ----
__global__ void pad_scatter_kernel(const float* __restrict__ hm,
                                   const int* __restrict__ centerHM,
                                   float* __restrict__ padded) {
  long long idx = (long long)blockIdx.x * blockDim.x + threadIdx.x;
  if (idx >= PAD_N) return;
  int x = (int)(idx % W_PAD);
  long long t = idx / W_PAD;
  int y = (int)(t % H_PAD); t /= H_PAD;
  int j = (int)(t % NJ);    t /= NJ;
  int v = (int)(t % NV);
  int b = (int)(t / NV);
  int cx = centerHM[(b * NV + v) * 2 + 0];
  int cy = centerHM[(b * NV + v) * 2 + 1];
  int c = x - (cx / 2 - HM / 2);
  int r = y - (cy / 2 - HM / 2);
  float val = 0.f;
  if ((unsigned)r < (unsigned)HM && (unsigned)c < (unsigned)HM)
    val = hm[((((long long)(b * NV + v)) * NJ + j) * HM + r) * HM + c];
  padded[idx] = val;
}

// ---- 2) prep: padded weight matrix [32][624] + K->Vol offset table ----
__global__ void prep_kernel(const float* __restrict__ w_v2v,
                            float* __restrict__ Wmat,
                            int* __restrict__ offTab) {
  int i = blockIdx.x * blockDim.x + threadIdx.x;
  if (i < CO_PAD * KPAD) {
    int co = i / KPAD, k = i % KPAD;
    Wmat[i] = (co < NJ && k < KTOT) ? w_v2v[co * KTOT + k] : 0.f;
  }
  if (i < KPAD) {
    if (i < KTOT) {
      int ci = i / 27, kk = i % 27;
      int kd = kk / 9, kh = (kk % 9) / 3, kw = kk % 3;
      offTab[i] = ci * (GP * GP * GZ) + kd * (GP * GZ) + kh * GZ + kw;
    } else {
      offTab[i] = 0;  // Wmat column is all-zero -> contribution vanishes
    }
  }
}

// ---- 3) reprojection gather: sample crops directly, sum views, scale 1/255,
//         write into zero-padded conv volume Vol[b][ci][x+1][y+1][z+1] ----
__global__ void reproject_gather_kernel(const float* __restrict__ hm,
                                        const int* __restrict__ centerHM,
                                        const float* __restrict__ center3D,
                                        const float* __restrict__ cam,   // [V,3,4]
                                        const float* __restrict__ Kin,   // [V,3,3]
                                        float* __restrict__ Vol) {
  int idx = blockIdx.x * blockDim.x + threadIdx.x;
  const int N = G * G * G;
  if (idx >= BB * N) return;
  int z = idx % G; int t = idx / G;
  int y = t % G;   t /= G;
  int x = t % G;   int b = t / G;
  float px = x * 8.f - 208.f + center3D[b * 3 + 0];
  float py = y * 8.f - 208.f + center3D[b * 3 + 1];
  float pz = z * 8.f - 208.f + center3D[b * 3 + 2];

  float acc[NJ];
#pragma unroll
  for (int j = 0; j < NJ; ++j) acc[j] = 0.f;

  for (int v = 0; v < NV; ++v) {
    const float* Kv = Kin + v * 9;
    const float* Cv = cam + v * 12;
    float P[3][4];
#pragma unroll
    for (int i = 0; i < 3; ++i)
#pragma unroll
      for (int c = 0; c < 4; ++c)
        P[i][c] = Kv[i * 3 + 0] * Cv[0 * 4 + c] + Kv[i * 3 + 1] * Cv[1 * 4 + c] +
                  Kv[i * 3 + 2] * Cv[2 * 4 + c];
    float p0 = P[0][0] * px + P[0][1] * py + P[0][2] * pz + P[0][3];
    float p1 = P[1][0] * px + P[1][1] * py + P[1][2] * pz + P[1][3];
    float p2 = P[2][0] * px + P[2][1] * py + P[2][2] * pz + P[2][3];
    float zc = (fabsf(p2) < 1e-6f) ? 1e-6f : p2;
    float u = p0 / zc, w = p1 / zc;
    // jnp.round == round-half-even == rintf (RTE default)
    int ui = (int)fminf(fmaxf(rintf(u * 0.5f), 0.f), (float)(W_PAD - 1));
    int vi = (int)fminf(fmaxf(rintf(w * 0.5f), 0.f), (float)(H_PAD - 1));
    int cx = centerHM[(b * NV + v) * 2 + 0];
    int cy = centerHM[(b * NV + v) * 2 + 1];
    int col = ui - (cx / 2 - HM / 2);
    int row = vi - (cy / 2 - HM / 2);
    if ((unsigned)row < (unsigned)HM && (unsigned)col < (unsigned)HM) {
      const float* base = hm + (((long long)(b * NV + v)) * NJ) * (HM * HM) + row * HM + col;
#pragma unroll
      for (int j = 0; j < NJ; ++j) acc[j] += base[(long long)j * (HM * HM)];
    }
  }
  float* vout = Vol + (long long)b * VOL_B + (long long)(x + 1) * (GP * GZ) + (y + 1) * GZ + (z + 1);
  const float inv255 = 1.f / 255.f;
#pragma unroll
  for (int j = 0; j < NJ; ++j) vout[(long long)j * VOL_CI] = acc[j] * inv255;
}

// ---- 4) V2V conv as implicit GEMM via V_WMMA_F32_16X16X4_F32, + bias + softplus ----
// One block = one (b,x,y) line. 8 waves: wave = (z_tile[0..3], m_tile[0..1]).
// Wave computes D[16 co x 16 z] accumulating over K=624 in steps of 4.
__global__ void conv_wmma_kernel(const float* __restrict__ Vol,
                                 const float* __restrict__ Wmat,
                                 const int* __restrict__ offTab,
                                 const float* __restrict__ bias,
                                 float* __restrict__ hf) {
  int bidx = blockIdx.x;
  int y = bidx % G; int t = bidx / G;
  int x = t % G;    int b = t / G;
  int wave   = threadIdx.x >> 5;
  int lane   = threadIdx.x & 31;
  int half   = lane >> 4;     // K-half selector for A/B fragments
  int lane16 = lane & 15;
  int m_tile = wave & 1;
  int z0     = (wave >> 1) * 16;   // 0,16,32,48 (z padded 52->64)

  int coA = m_tile * 16 + lane16;  // A-matrix row held by this lane
  long long baseB = (long long)b * VOL_B + (long long)x * (GP * GZ) + (long long)y * GZ +
                    (z0 + lane16);

  v8f c = {0.f, 0.f, 0.f, 0.f, 0.f, 0.f, 0.f, 0.f};
  for (int kb = 0; kb < KPAD; kb += 4) {
    int k0 = kb + 2 * half;
    v2f a, bv;
    a.x  = Wmat[coA * KPAD + k0];
    a.y  = Wmat[coA * KPAD + k0 + 1];
    bv.x = Vol[baseB + offTab[k0]];
    bv.y = Vol[baseB + offTab[k0 + 1]];
    c = __builtin_amdgcn_wmma_f32_16x16x4_f32(false, a, false, bv, (short)0, c,
                                              false, false);
  }

  int z = z0 + lane16;
  if (z < G) {
#pragma unroll
    for (int r = 0; r < 8; ++r) {
      int co = m_tile * 16 + r + 8 * half;   // C/D layout: M = r + 8*(lane>=16)
      if (co < NJ) {
        float vv = c[r] + bias[co];
        float sp = fmaxf(vv, 0.f) + log1pf(expf(-fabsf(vv)));  // stable softplus
        hf[((((long long)b * NJ + co) * G + x) * G + y) * G + z] = sp;
      }
    }
  }
}

// ---- 5) soft-argmax reduction: one block per (b,j) ----
__global__ void softargmax_kernel(const float* __restrict__ hf,
                                  const float* __restrict__ center3D,
                                  float* __restrict__ pts) {
  int bj = blockIdx.x;
  int b = bj / NJ;
  const float* p = hf + (long long)bj * (G * G * G);
  float s = 0.f, sx = 0.f, sy = 0.f, sz = 0.f;
  for (int i = threadIdx.x; i < G * G * G; i += blockDim.x) {
    int z = i % G; int t = i / G; int y = t % G; int x = t / G;
    float v = p[i];
    s += v; sx += v * x; sy += v * y; sz += v * z;
  }
  __shared__ float red[4][256];
  red[0][threadIdx.x] = s;  red[1][threadIdx.x] = sx;
  red[2][threadIdx.x] = sy; red[3][threadIdx.x] = sz;
  __syncthreads();
  for (int off = 128; off > 0; off >>= 1) {
    if ((int)threadIdx.x < off) {
#pragma unroll
      for (int q = 0; q < 4; ++q) red[q][threadIdx.x] += red[q][threadIdx.x + off];
    }
    __syncthreads();
  }
  if (threadIdx.x == 0) {
    float n = red[0][0];
    pts[bj * 3 + 0] = (red[1][0] / n) * 8.f - 208.f + center3D[b * 3 + 0];
    pts[bj * 3 + 1] = (red[2][0] / n) * 8.f - 208.f + center3D[b * 3 + 1];
    pts[bj * 3 + 2] = (red[3][0] / n) * 8.f - 208.f + center3D[b * 3 + 2];
  }
}

extern "C" void kernel_launch(void* const* d_in, const int* in_sizes, int n_in,
                              void* d_out, int out_size, void* d_ws, size_t ws_size,
                              hipStream_t stream) {
  const float* heatmaps = (const float*)d_in[0];
  const int*   centerHM = (const int*)d_in[1];
  const float* center3D = (const float*)d_in[2];
  const float* cam      = (const float*)d_in[3];
  const float* Kin      = (const float*)d_in[4];
  const float* w_v2v    = (const float*)d_in[6];
  const float* b_v2v    = (const float*)d_in[7];

  float* out    = (float*)d_out;
  float* hf     = out;                 // [B,J,G,G,G]
  float* padded = out + HF_N;          // [B,V,J,512,640]
  float* pts    = out + HF_N + PAD_N;  // [B,J,3]

  float* Vol  = (float*)d_ws;                    // [B,23,54,54,72] zero-padded
  float* Wmat = Vol + VOL_N;                     // [32,624]
  int* offTab = (int*)(Wmat + CO_PAD * KPAD);    // [624]

  hipMemsetAsync(Vol, 0, (size_t)VOL_N * sizeof(float), stream);

  prep_kernel<<<(CO_PAD * KPAD + 255) / 256, 256, 0, stream>>>(w_v2v, Wmat, offTab);

  pad_scatter_kernel<<<(unsigned)((PAD_N + 255) / 256), 256, 0, stream>>>(
      heatmaps, centerHM, padded);

  int BN = BB * G * G * G;
  reproject_gather_kernel<<<(BN + 255) / 256, 256, 0, stream>>>(
      heatmaps, centerHM, center3D, cam, Kin, Vol);

  conv_wmma_kernel<<<BB * G * G, 256, 0, stream>>>(Vol, Wmat, offTab, b_v2v, hf);

  softargmax_kernel<<<BB * NJ, 256, 0, stream>>>(hf, center3D, pts);
}